// CannyLayer_49993419325666
// MI455X (gfx1250) — compile-verified
//
#include <hip/hip_runtime.h>

typedef __attribute__((ext_vector_type(2))) float v2f;
typedef __attribute__((ext_vector_type(8))) float v8f;

#define B_   32
#define H_   512
#define W_   512
#define NPX  (B_*H_*W_)      // 8388608 pixels
#define NW   (NPX/32)        // 262144 bit-words
#define WPR  (W_/32)         // 16 words per row

// ---------------------------------------------------------------------------
// Branch-free Gaussian band weight: G[dy][d]/159, 0 outside [0,4].
// Exploits double symmetry (G[dy][d] == G[4-dy][d] == G[dy][4-d]); compiles
// to a short v_cndmask chain — no memory, no exec-mask divergence.
// ---------------------------------------------------------------------------
__device__ __forceinline__ float gweight(int dy, int d) {
  bool in = (unsigned)d <= 4u;
  int r = (dy < 3) ? dy : 4 - dy;           // 0..2
  int c = (d  < 3) ? d  : 4 - d;            // 0..2 (junk if !in, masked below)
  float v =
      (r == 0) ? ((c == 0) ? 2.f  : (c == 1) ? 4.f  : 5.f)  :
      (r == 1) ? ((c == 0) ? 4.f  : (c == 1) ? 9.f  : 12.f) :
                 ((c == 0) ? 5.f  : (c == 1) ? 12.f : 15.f);
  return in ? v * (1.f / 159.f) : 0.f;
}

// ---------------------------------------------------------------------------
// 1) 5x5 Gaussian blur as 25 x V_WMMA_F32_16X16X4_F32 per 16x16 tile.
//    out(16x16) = sum_{dy=0..4} A_dy(16x20) * W_dy(20x16), W banded with G[dy].
//    One wave per tile; 20x20 zero-padded patch staged in LDS.
// ---------------------------------------------------------------------------
__global__ __launch_bounds__(256) void canny_blur_wmma(const float* __restrict__ img,
                                                       float* __restrict__ sm) {
  __shared__ float patch[8 * 400];          // 8 waves x 20x20 floats
  const int lane = threadIdx.x & 31;
  const int wave = threadIdx.x >> 5;
  const int tile = blockIdx.x * 8 + wave;   // 32768 tiles total
  const int b  = tile >> 10;
  const int ty = (tile >> 5) & 31;
  const int tx = tile & 31;
  const int y0 = ty * 16, x0 = tx * 16;
  const float* ib = img + (size_t)b * H_ * W_;
  float* lp = patch + wave * 400;

  // load 20x20 halo patch (zero pad at image borders)
  for (int idx = lane; idx < 400; idx += 32) {
    int r = idx / 20, c = idx - r * 20;
    int gy = y0 + r - 2, gx = x0 + c - 2;
    float v = 0.f;
    if ((unsigned)gy < (unsigned)H_ && (unsigned)gx < (unsigned)W_)
      v = ib[gy * W_ + gx];
    lp[idx] = v;
  }
  __syncthreads();

  const int hh   = lane >> 4;   // half-wave select (K offset 0 or 2)
  const int mrow = lane & 15;   // A: row M; B: col N

  v8f acc = {};
#pragma unroll
  for (int dy = 0; dy < 5; ++dy) {
#pragma unroll
    for (int kb = 0; kb < 5; ++kb) {
      const int c0 = kb * 4 + 2 * hh;       // contraction index for .x
      v2f a, w;
      // A 16x4 f32 layout: lanes 0-15 K={0,1}, lanes 16-31 K={2,3}
      a.x = lp[(mrow + dy) * 20 + c0];
      a.y = lp[(mrow + dy) * 20 + c0 + 1];
      // B 4x16 banded: W[c][n] = G[dy][c-n] for 0<=c-n<=4 (select chain, no mem)
      w.x = gweight(dy, c0 - mrow);
      w.y = gweight(dy, c0 + 1 - mrow);
      acc = __builtin_amdgcn_wmma_f32_16x16x4_f32(
          /*neg_a=*/false, a, /*neg_b=*/false, w,
          /*c_mod=*/(short)0, acc, /*reuse_a=*/false, /*reuse_b=*/false);
    }
  }

  // C/D layout: VGPR r holds rows r (lanes 0-15) and r+8 (lanes 16-31)
  float* ob = sm + (size_t)b * H_ * W_;
#pragma unroll
  for (int r = 0; r < 8; ++r) {
    int m = r + 8 * hh;
    ob[(y0 + m) * W_ + x0 + mrow] = acc[r];
  }
}

// ---------------------------------------------------------------------------
// 2) Sobel gradient, magnitude, quantized direction (no atan2 needed).
// ---------------------------------------------------------------------------
__global__ __launch_bounds__(256) void canny_grad(const float* __restrict__ sm,
                                                  float* __restrict__ mag,
                                                  unsigned char* __restrict__ dir) {
  int idx = blockIdx.x * 256 + threadIdx.x;
  if (idx >= NPX) return;
  int x = idx & (W_ - 1);
  int y = (idx >> 9) & (H_ - 1);
  int b = idx >> 18;
  const float* sb = sm + (size_t)b * H_ * W_;

  float v[3][3];
#pragma unroll
  for (int di = 0; di < 3; ++di)
#pragma unroll
    for (int dj = 0; dj < 3; ++dj) {
      int yy = y + di - 1, xx = x + dj - 1;
      v[di][dj] = ((unsigned)yy < (unsigned)H_ && (unsigned)xx < (unsigned)W_)
                      ? sb[yy * W_ + xx] : 0.f;
    }
  float gx = (v[0][2] - v[0][0]) + 2.f * (v[1][2] - v[1][0]) + (v[2][2] - v[2][0]);
  float gy = (v[2][0] - v[0][0]) + 2.f * (v[2][1] - v[0][1]) + (v[2][2] - v[0][2]);
  float m = sqrtf(gx * gx + gy * gy);

  float ax = fabsf(gx), ay = fabsf(gy);
  int d;
  if (ay >= 2.41421356f * ax)      d = 2;                 // 90 deg
  else if (ay < 0.41421356f * ax)  d = 0;                 // 0 deg
  else                             d = (gx * gy >= 0.f) ? 1 : 3; // 45 / 135

  mag[idx] = m;
  dir[idx] = (unsigned char)d;
}

// ---------------------------------------------------------------------------
// 3) NMS + double threshold; bit-pack strong/weak masks via wave32 ballot.
// ---------------------------------------------------------------------------
__device__ __forceinline__ float mag_at(const float* mb, int y, int x) {
  if ((unsigned)y < (unsigned)H_ && (unsigned)x < (unsigned)W_)
    return mb[y * W_ + x];
  return 0.f;
}

__global__ __launch_bounds__(256) void canny_nms(const float* __restrict__ mag,
                                                 const unsigned char* __restrict__ dir,
                                                 float* __restrict__ out_nms,
                                                 unsigned int* __restrict__ strongb,
                                                 unsigned int* __restrict__ weakb) {
  int idx = blockIdx.x * 256 + threadIdx.x;
  if (idx >= NPX) return;
  int x = idx & (W_ - 1);
  int y = (idx >> 9) & (H_ - 1);
  int b = idx >> 18;
  const float* mb = mag + (size_t)b * H_ * W_;

  float m = mb[y * W_ + x];
  int d = dir[idx];
  int dy1, dx1, dy2, dx2;
  switch (d) {
    case 0:  dy1 = 0;  dx1 = 1;  dy2 = 0;  dx2 = -1; break;
    case 1:  dy1 = -1; dx1 = 1;  dy2 = 1;  dx2 = -1; break;
    case 2:  dy1 = 1;  dx1 = 0;  dy2 = -1; dx2 = 0;  break;
    default: dy1 = -1; dx1 = -1; dy2 = 1;  dx2 = 1;  break;
  }
  float n1 = mag_at(mb, y + dy1, x + dx1);
  float n2 = mag_at(mb, y + dy2, x + dx2);
  float nms = (m >= n1 && m >= n2) ? m : 0.f;

  out_nms[idx] = nms;

  unsigned int smask = __builtin_amdgcn_ballot_w32(nms > 0.3f);
  unsigned int wmask = __builtin_amdgcn_ballot_w32(nms > 0.1f);
  if ((threadIdx.x & 31) == 0) {
    strongb[idx >> 5] = smask;
    weakb[idx >> 5]   = wmask;
  }
}

// ---------------------------------------------------------------------------
// 4) One hysteresis step on bit-packed masks: 3x3 dilation AND weak.
// ---------------------------------------------------------------------------
__global__ __launch_bounds__(256) void canny_hyst(const unsigned int* __restrict__ cur,
                                                  const unsigned int* __restrict__ weak,
                                                  unsigned int* __restrict__ nxt) {
  int w = blockIdx.x * 256 + threadIdx.x;
  if (w >= NW) return;
  int wx = w & (WPR - 1);
  int y  = (w >> 4) & (H_ - 1);

  unsigned int dil = 0u;
#pragma unroll
  for (int dy = -1; dy <= 1; ++dy) {
    int yy = y + dy;
    if ((unsigned)yy < (unsigned)H_) {
      int base = w + dy * WPR;
      unsigned int c = cur[base];
      unsigned int l = (wx > 0)        ? cur[base - 1] : 0u;
      unsigned int r = (wx < WPR - 1)  ? cur[base + 1] : 0u;
      dil |= c | (c << 1) | (c >> 1) | (l >> 31) | (r << 31);
    }
  }
  nxt[w] = weak[w] & dil;
}

// ---------------------------------------------------------------------------
// 5) Final mask: zero d_out where final strong bit is unset.
// ---------------------------------------------------------------------------
__global__ __launch_bounds__(256) void canny_final(const unsigned int* __restrict__ bits,
                                                   float* __restrict__ out) {
  int idx = blockIdx.x * 256 + threadIdx.x;
  if (idx >= NPX) return;
  unsigned int wd = bits[idx >> 5];
  if (!((wd >> (idx & 31)) & 1u)) out[idx] = 0.f;
}

// ---------------------------------------------------------------------------
extern "C" void kernel_launch(void* const* d_in, const int* in_sizes, int n_in,
                              void* d_out, int out_size, void* d_ws, size_t ws_size,
                              hipStream_t stream) {
  (void)in_sizes; (void)n_in; (void)out_size; (void)ws_size;
  const float* img = (const float*)d_in[0];
  float* out = (float*)d_out;

  char* ws = (char*)d_ws;
  float*         sm    = (float*)ws;                                 // NPX f32
  float*         mag   = (float*)(ws + (size_t)NPX * 4);             // NPX f32
  unsigned char* dir   = (unsigned char*)(ws + (size_t)NPX * 8);     // NPX u8
  unsigned int*  bitsA = (unsigned int*)(ws + (size_t)NPX * 9);      // NW u32
  unsigned int*  bitsB = bitsA + NW;                                 // NW u32
  unsigned int*  weakb = bitsB + NW;                                 // NW u32

  canny_blur_wmma<<<4096, 256, 0, stream>>>(img, sm);
  canny_grad<<<NPX / 256, 256, 0, stream>>>(sm, mag, dir);
  canny_nms<<<NPX / 256, 256, 0, stream>>>(mag, dir, out, bitsA, weakb);

  unsigned int* cur = bitsA;
  unsigned int* nxt = bitsB;
  for (int it = 0; it < 32; ++it) {
    canny_hyst<<<NW / 256, 256, 0, stream>>>(cur, weakb, nxt);
    unsigned int* t = cur; cur = nxt; nxt = t;
  }
  canny_final<<<NPX / 256, 256, 0, stream>>>(cur, out);
}